// GCN_80358838108791
// MI455X (gfx1250) — compile-verified
//
#include <hip/hip_runtime.h>
#include <math.h>

// Problem constants (reference: B=64, C=64, D=512, O=256, 8 regions of 8 channels)
#define BDIM 64
#define CDIM 64
#define DDIM 512
#define ODIM 256
#define NREG 8
#define GSZ  8
#define MROWS (BDIM * CDIM)   // 4096 rows for all dense GEMMs
#define KDIM DDIM             // all GEMMs have K = 512
#define KPAD 516              // LDS row pitch (pad 4) -> conflict-free b64 reads

typedef __attribute__((ext_vector_type(2))) float v2f;
typedef __attribute__((ext_vector_type(8))) float v8f;

typedef __attribute__((address_space(1))) int* gint_p;   // global int*
typedef __attribute__((address_space(3))) int* lint_p;   // LDS int*

#if defined(__has_builtin)
#if __has_builtin(__builtin_amdgcn_global_load_async_to_lds_b32) && \
    __has_builtin(__builtin_amdgcn_s_wait_asynccnt)
#define HAVE_ASYNC_LDS 1
#endif
#endif

// ---------------------------------------------------------------------------
// FP32 WMMA GEMM:  Y[M,N] = epilogue(A[M,512] @ W[512,N] + bias)
// One wave computes one 16x16 output tile via V_WMMA_F32_16X16X4_F32.
// Block = 8 waves = 8 M-tiles sharing one N-tile; the shared 512x16 W slice is
// staged once into LDS (transposed, padded) via gfx1250 async-to-LDS copies so
// the inner loop is: global b64 (A) + ds b64 (B) + wmma.
// MODE: 0 = none; 1 = +bias; 2 = 0.5*leaky(.+bias)+0.5*R; 3 = leaky(.+bias)
// ---------------------------------------------------------------------------
template <int MODE>
__global__ __launch_bounds__(256) void gemm_wmma_f32(
    const float* __restrict__ A, const float* __restrict__ W,
    const float* __restrict__ bias, const float* __restrict__ R,
    float* __restrict__ Y, int N)
{
    __shared__ __align__(16) float Wt[16 * KPAD];   // Wt[n][k], pitch KPAD

    const int tid = threadIdx.x;
    const int n0 = blockIdx.x * 16;

    // ---- stage W[:, n0:n0+16] into LDS, transposed ----
    for (int e = tid; e < 16 * KDIM; e += 256) {
        const int n = e & 15;        // consecutive lanes -> consecutive n (coalesced)
        const int k = e >> 4;
#ifdef HAVE_ASYNC_LDS
        __builtin_amdgcn_global_load_async_to_lds_b32(
            (gint_p)(W + (size_t)k * N + n0 + n),
            (lint_p)(Wt + n * KPAD + k),
            /*offset=*/0, /*cpol=*/0);
#else
        Wt[n * KPAD + k] = W[(size_t)k * N + n0 + n];
#endif
    }
#ifdef HAVE_ASYNC_LDS
    __builtin_amdgcn_s_wait_asynccnt(0);
#endif
    __syncthreads();

    const int lane = tid & 31;
    const int wave = tid >> 5;
    const int lo = lane & 15;        // row within M-tile / col within N-tile
    const int hi = lane >> 4;        // selects K pair {0,1} vs {2,3}
    const int m0 = (blockIdx.y * 8 + wave) * 16;

    const float* Arow = A + (size_t)(m0 + lo) * KDIM;   // this lane's A row
    const float* Bcol = Wt + lo * KPAD;                 // this lane's W column in LDS

    v8f acc = {};
#pragma unroll 4
    for (int k = 0; k < KDIM; k += 4) {
        const int ka = k + 2 * hi;
        v2f af = *(const v2f*)(Arow + ka);   // global b64
        v2f bf = *(const v2f*)(Bcol + ka);   // ds b64, bank-conflict-free
        acc = __builtin_amdgcn_wmma_f32_16x16x4_f32(
            /*neg_a=*/false, af, /*neg_b=*/false, bf,
            /*c_mod=*/(short)0, acc, /*reuse_a=*/false, /*reuse_b=*/false);
    }

    const int n = n0 + lo;
    const float bv = (MODE >= 1) ? bias[n] : 0.0f;
#pragma unroll
    for (int v = 0; v < 8; ++v) {
        const int m = m0 + v + 8 * hi;      // C/D layout: VGPR v -> row v + 8*hi
        float val = acc[v] + bv;
        if (MODE == 2 || MODE == 3) val = (val >= 0.0f) ? val : 0.2f * val;
        if (MODE == 2) val = 0.5f * val + 0.5f * R[(size_t)m * N + n];
        Y[(size_t)m * N + n] = val;
    }
}

// ---------------------------------------------------------------------------
// Per (batch, region) block: 8x8 attention scores (fused ReLU + dot with W2),
// masked softmax (diag excluded; b2 dropped by shift-invariance; b1 folded
// into bp's GEMM bias), then m1 = adj @ x for this region's 8 rows.
// ---------------------------------------------------------------------------
__global__ __launch_bounds__(256) void region_attn(
    const float* __restrict__ a, const float* __restrict__ bp,
    const float* __restrict__ W2, const float* __restrict__ x,
    const int* __restrict__ regidx,
    float* __restrict__ adj_out, float* __restrict__ m1)
{
    __shared__ float w2s[DDIM];
    __shared__ int   ch[GSZ];
    __shared__ float sc[GSZ][GSZ];
    __shared__ float adjs[GSZ][GSZ];

    const int tid = threadIdx.x;
    const int blk = blockIdx.x;        // b*NREG + r
    const int b = blk >> 3;
    const int r = blk & 7;

    for (int d = tid; d < DDIM; d += 256) w2s[d] = W2[d];
    if (tid < GSZ) ch[tid] = regidx[r * GSZ + tid];
    __syncthreads();

    // ---- scores: 64 pairs x 4 threads each ----
    const int pair = tid >> 2;
    const int sub  = tid & 3;
    const int i = pair >> 3, j = pair & 7;
    const float* ai = a  + (size_t)(b * CDIM + ch[i]) * DDIM;
    const float* bj = bp + (size_t)(b * CDIM + ch[j]) * DDIM;
    float s = 0.0f;
    for (int d = sub; d < DDIM; d += 4) {
        float h = fmaxf(ai[d] + bj[d], 0.0f);   // relu(a_i + bp_j + b1)
        s += h * w2s[d];
    }
    s += __shfl_xor(s, 1, 32);
    s += __shfl_xor(s, 2, 32);
    if (sub == 0) sc[i][j] = s;
    __syncthreads();

    // ---- masked softmax per row (7 valid entries, diag = 0) ----
    if (tid < GSZ) {
        const int row = tid;
        float mx = -INFINITY;
        for (int jj = 0; jj < GSZ; ++jj)
            if (jj != row) mx = fmaxf(mx, sc[row][jj]);
        float e[GSZ], sum = 0.0f;
        for (int jj = 0; jj < GSZ; ++jj) {
            e[jj] = (jj == row) ? 0.0f : expf(sc[row][jj] - mx);
            sum += e[jj];
        }
        const float inv = 1.0f / sum;
        for (int jj = 0; jj < GSZ; ++jj) {
            const float v = e[jj] * inv;
            adjs[row][jj] = v;
            adj_out[(size_t)blk * 64 + row * GSZ + jj] = v;
        }
    }
    __syncthreads();

    // ---- m1[ch[i], :] = sum_j adj[i][j] * x[ch[j], :] ----
    for (int idx = tid; idx < GSZ * DDIM; idx += 256) {
        const int row = idx >> 9;            // DDIM == 512
        const int d   = idx & (DDIM - 1);
        float sum = 0.0f;
#pragma unroll
        for (int jj = 0; jj < GSZ; ++jj)
            sum += adjs[row][jj] * x[(size_t)(b * CDIM + ch[jj]) * DDIM + d];
        m1[(size_t)(b * CDIM + ch[row]) * DDIM + d] = sum;
    }
}

// ---------------------------------------------------------------------------
// m2 = adj @ x1 (per region, adj read back from workspace)
// ---------------------------------------------------------------------------
__global__ __launch_bounds__(256) void region_aggregate(
    const float* __restrict__ adj_in, const float* __restrict__ xin,
    const int* __restrict__ regidx, float* __restrict__ mout)
{
    __shared__ int   ch[GSZ];
    __shared__ float adjs[GSZ][GSZ];
    const int tid = threadIdx.x;
    const int blk = blockIdx.x;
    const int b = blk >> 3;
    const int r = blk & 7;
    if (tid < GSZ) ch[tid] = regidx[r * GSZ + tid];
    if (tid < 64) adjs[tid >> 3][tid & 7] = adj_in[(size_t)blk * 64 + tid];
    __syncthreads();
    for (int idx = tid; idx < GSZ * DDIM; idx += 256) {
        const int row = idx >> 9;
        const int d   = idx & (DDIM - 1);
        float sum = 0.0f;
#pragma unroll
        for (int jj = 0; jj < GSZ; ++jj)
            sum += adjs[row][jj] * xin[(size_t)(b * CDIM + ch[jj]) * DDIM + d];
        mout[(size_t)(b * CDIM + ch[row]) * DDIM + d] = sum;
    }
}

// ---------------------------------------------------------------------------
extern "C" void kernel_launch(void* const* d_in, const int* in_sizes, int n_in,
                              void* d_out, int out_size, void* d_ws, size_t ws_size,
                              hipStream_t stream)
{
    (void)in_sizes; (void)n_in; (void)out_size; (void)ws_size;

    const float* x    = (const float*)d_in[0];   // [B,C,D]
    const float* W1   = (const float*)d_in[1];   // [2D,D]
    const float* b1   = (const float*)d_in[2];   // [D]
    const float* W2   = (const float*)d_in[3];   // [D,1]
    /* b2 = d_in[4] unused: softmax is shift-invariant */
    const float* Wg1  = (const float*)d_in[5];   // [D,D]
    const float* bg1  = (const float*)d_in[6];   // [D]
    const float* Wg2  = (const float*)d_in[7];   // [D,O]
    const float* bg2  = (const float*)d_in[8];   // [O]
    const int* regidx = (const int*)d_in[9];     // [NREG,GSZ]
    float* out = (float*)d_out;                  // [B,C,O]

    float* ws = (float*)d_ws;
    const size_t SZ = (size_t)MROWS * DDIM;      // 2M floats per [B,C,D] buffer
    float* a_   = ws;            // x @ W1[:D]
    float* bp_  = ws + 1 * SZ;   // x @ W1[D:] + b1
    float* m1_  = ws + 2 * SZ;
    float* x1_  = ws + 3 * SZ;
    float* adj_ = ws + 4 * SZ;   // [B*NREG][8][8] = 32K floats
    float* m2_  = a_;            // a_ dead after region_attn -> reuse

    dim3 blk(256);
    dim3 g512(DDIM / 16, MROWS / 128);   // (32, 32)
    dim3 g256(ODIM / 16, MROWS / 128);   // (16, 32)
    dim3 greg(BDIM * NREG);              // 512 region blocks

    // 1) a = x @ W1[:D]
    gemm_wmma_f32<0><<<g512, blk, 0, stream>>>(x, W1, nullptr, nullptr, a_, DDIM);
    // 2) bp = x @ W1[D:] + b1
    gemm_wmma_f32<1><<<g512, blk, 0, stream>>>(x, W1 + (size_t)DDIM * DDIM, b1, nullptr, bp_, DDIM);
    // 3) masked attention + softmax + m1 = adj @ x
    region_attn<<<greg, blk, 0, stream>>>(a_, bp_, W2, x, regidx, adj_, m1_);
    // 4) x1 = 0.5*leaky(m1 @ Wg1 + bg1) + 0.5*x
    gemm_wmma_f32<2><<<g512, blk, 0, stream>>>(m1_, Wg1, bg1, x, x1_, DDIM);
    // 5) m2 = adj @ x1
    region_aggregate<<<greg, blk, 0, stream>>>(adj_, x1_, regidx, m2_);
    // 6) out = leaky(m2 @ Wg2 + bg2)
    gemm_wmma_f32<3><<<g256, blk, 0, stream>>>(m2_, Wg2, bg2, nullptr, out, ODIM);
}